// CppFUSEModel_5566277616490
// MI455X (gfx1250) — compile-verified
//
#include <hip/hip_runtime.h>
#include <math.h>

// ---------------------------------------------------------------------------
// FUSE hydrological model on MI455X (gfx1250, wave32)
//   Phase 1: sequential scan, 2 ensembles per wave32 (16 lanes per ensemble),
//            NB-reductions via intra-wave shuffles, runoff -> d_ws.
//            __powf on [0,1] states keeps the 8192-long dependent chain short.
//   Phase 2: UH routing as per-ensemble Toeplitz GEMM using
//            V_WMMA_F32_16X16X4_F32 (full fp32 precision), 16 chained WMMAs
//            per 256-timestep output tile. UH weights stored REVERSED and
//            zero-padded so each B fragment is one ascending-order
//            ds_load_2addr_b32 straight into the WMMA operand pair.
// ---------------------------------------------------------------------------

typedef __attribute__((ext_vector_type(2))) float v2f;
typedef __attribute__((ext_vector_type(8))) float v8f;

#define UH_LEN 30
#define TILE_T 256    // outputs per WMMA tile (16x16)
#define WIN    304    // im2col window: 48 left halo + 256
#define WIN_PAD 320
#define WPAD   80     // reversed padded UH weights: rw[i] = w[63-i], else 0

__device__ __forceinline__ float sigmoid_scale(float x, float lo, float hi) {
    return lo + (hi - lo) / (1.0f + __expf(-x));
}

// ------------------------------ Phase 1 ------------------------------------
// grid = ceil(E/2) blocks of 32 threads. Lane layout: lanes 0-15 = ensemble
// 2*bid, lanes 16-31 = ensemble 2*bid+1; within each half, lane&15 = band.
__global__ __launch_bounds__(32) void fuse_scan_kernel(
    const float* __restrict__ raw,      // [E,29]
    const float* __restrict__ forcing,  // [T,3]
    const float* __restrict__ st0,      // [2]
    const float* __restrict__ areaf,    // [NB]
    const float* __restrict__ elev,     // [NB]
    float* __restrict__ runoff,         // [E,T] (workspace)
    int T, int E)
{
    const int lane = threadIdx.x;
    const int nb   = lane & 15;
    const int e    = blockIdx.x * 2 + (lane >> 4);
    if (e >= E) return;   // whole 16-lane half exits together

    const float* rp = raw + (size_t)e * 29;
    // physical parameters (sigmoid-rescaled), indices per PARAM_NAMES
    const float s1max = sigmoid_scale(rp[0],  50.0f,   5000.0f);
    const float s2max = sigmoid_scale(rp[1],  100.0f, 10000.0f);
    const float ku    = sigmoid_scale(rp[6],  0.01f,   1000.0f);
    const float cexp  = sigmoid_scale(rp[7],  1.0f,      20.0f);
    const float ki    = sigmoid_scale(rp[11], 0.01f,   1000.0f);
    const float ks    = sigmoid_scale(rp[12], 0.001f, 10000.0f);
    const float nexp  = sigmoid_scale(rp[13], 1.0f,      10.0f);
    const float acmax = sigmoid_scale(rp[17], 0.05f,     0.95f);
    const float bexp  = sigmoid_scale(rp[18], 0.001f,     3.0f);
    const float train = sigmoid_scale(rp[22], -2.0f,      4.0f);
    const float tmelt = sigmoid_scale(rp[23], -2.0f,      4.0f);
    const float mrate = sigmoid_scale(rp[24], 1.0f,      10.0f);
    const float lapse = sigmoid_scale(rp[25], -9.8f,      0.0f);
    const float opg   = sigmoid_scale(rp[26], 0.0f,       1.0f);

    const float delev  = (elev[nb] - 1500.0f) * 0.001f;
    const float tshift = lapse * delev;
    const float pmult  = fmaxf(1.0f + opg * delev, 0.0f);
    const float af     = areaf[nb];

    float swe = 0.0f;
    float s1  = st0[0];
    float s2  = st0[1];

    for (int t = 0; t < T; ++t) {
        // wave-uniform forcing read; prefetch ahead into L2 (global_prefetch)
        const float p    = forcing[3 * t + 0];
        const float pet  = forcing[3 * t + 1];
        const float tair = forcing[3 * t + 2];
        __builtin_prefetch(forcing + 3 * ((t + 96 < T) ? (t + 96) : t), 0, 0);

        const float tb   = tair + tshift;
        const float pb   = p * pmult;
        const float rain = (tb > train) ? pb : 0.0f;
        const float snow = pb - rain;
        const float melt = fminf(mrate * fmaxf(tb - tmelt, 0.0f), swe + snow);
        swe = swe + snow - melt;
        const float infl = rain + melt;

        // w1, w2 are in [0,1] by construction: fast __powf is safe here and
        // keeps the dependent chain short (v_log/v_mul/v_exp, no branches).
        const float w1   = fminf(fmaxf(s1 / s1max, 0.0f), 1.0f);
        const float evap = pet * w1;
        const float qsx  = acmax * __powf(w1, bexp) * infl;
        const float perc = ku * __powf(w1, cexp);
        const float qif  = ki * w1;
        const float s1n  = s1 + infl - qsx - evap - perc - qif;
        const float ovr1 = fmaxf(s1n - s1max, 0.0f);
        s1 = fminf(fmaxf(s1n, 0.0f), s1max);
        const float qsurf = qsx + ovr1;

        // recharge: reduce over 16 bands (butterfly stays in the half-wave)
        float rech = af * perc;
        rech += __shfl_xor(rech, 1, 32);
        rech += __shfl_xor(rech, 2, 32);
        rech += __shfl_xor(rech, 4, 32);
        rech += __shfl_xor(rech, 8, 32);

        const float w2   = fminf(fmaxf(s2 / s2max, 0.0f), 1.0f);
        const float qb   = ks * __powf(w2, nexp);
        const float s2n  = s2 + rech - qb;
        const float ovr2 = fmaxf(s2n - s2max, 0.0f);
        s2 = fminf(fmaxf(s2n, 0.0f), s2max);

        float qs = af * (qsurf + qif);
        qs += __shfl_xor(qs, 1, 32);
        qs += __shfl_xor(qs, 2, 32);
        qs += __shfl_xor(qs, 4, 32);
        qs += __shfl_xor(qs, 8, 32);

        if (nb == 0) runoff[(size_t)e * T + t] = qs + qb + ovr2;
    }
}

// ------------------------------ Phase 2 ------------------------------------
// One wave32 per (ensemble, 256-output tile).
//   D[m,n] = routed[t0 + 16m + n] = sum_{k=0}^{63} A[m,k] * B[k,n]
//   A[m,k] = runoff[e, t0 - 48 + 16m + k]          (im2col window in LDS)
//   B[k,n] = w[e, n + 48 - k]  (zero outside lag [0,30))
// 16 chained V_WMMA_F32_16X16X4_F32 cover K=64 at full f32 precision.
// Weights are stored reversed (rw[i] = w[63-i]) so B fragments load as
// ascending-adjacent pairs: one ds_load_2addr_b32, no operand swaps.
__global__ __launch_bounds__(32) void fuse_route_wmma_kernel(
    const float* __restrict__ raw,      // [E,29] (mu_t at col 21)
    const float* __restrict__ runoff,   // [E,T]
    float* __restrict__ out,            // [E,T]
    int T, int tilesPerE)
{
    __shared__ float win[WIN_PAD];
    __shared__ float rw[WPAD];          // rw[63 - lag] = w[lag], else 0

    const int lane = threadIdx.x;
    const int e    = blockIdx.x / tilesPerE;
    const int tile = blockIdx.x % tilesPerE;
    const int t0   = tile * TILE_T;

    // ---- gamma(2.5, delay) UH weights; lgamma & delay^k cancel in normalize
    const float delay = sigmoid_scale(raw[(size_t)e * 29 + 21], 0.01f, 5.0f);
    const float tmid  = (float)lane + 0.5f;
    float u = (lane < UH_LEN) ? __powf(tmid, 1.5f) * __expf(-tmid / delay) : 0.0f;
    float s = u;
    s += __shfl_xor(s, 16, 32);
    s += __shfl_xor(s, 8, 32);
    s += __shfl_xor(s, 4, 32);
    s += __shfl_xor(s, 2, 32);
    s += __shfl_xor(s, 1, 32);

    // zero-fill padded reversed weights, then scatter w[lane] at 63-lane
    rw[lane]      = 0.0f;
    rw[lane + 32] = 0.0f;
    if (lane < WPAD - 64) rw[lane + 64] = 0.0f;
    __syncthreads();
    if (lane < UH_LEN) rw[63 - lane] = u / s;

    // ---- stage im2col window: runoff[e, t0-48 .. t0+255], zero left halo
    for (int i = lane; i < WIN_PAD; i += 32) {
        const int t = t0 - 48 + i;
        float v = 0.0f;
        if (i < WIN && t >= 0) v = runoff[(size_t)e * T + t];
        win[i] = v;
    }
    __syncthreads();

    // ---- fragment coordinates (wave32 layouts, ISA 7.12.2)
    const int nm = lane & 15;       // A: row M / B: col N
    const int hi = lane >> 4;       // VGPR half: K base 0 (lo) / 2 (hi)

    v8f c = {0.f, 0.f, 0.f, 0.f, 0.f, 0.f, 0.f, 0.f};
#pragma unroll
    for (int kc = 0; kc < 16; ++kc) {
        const int k0 = 4 * kc + 2 * hi;     // this lane's K for VGPR0
        v2f a, b;
        a.x = win[16 * nm + k0];            // A[M, k0]
        a.y = win[16 * nm + k0 + 1];        // A[M, k0+1]
        const int i0 = 15 - nm + k0;        // rw index: ascending pair
        b.x = rw[i0];                       // = w[nm+48-k0] = B[k0,   N]
        b.y = rw[i0 + 1];                   // = w[nm+47-k0] = B[k0+1, N]
        c = __builtin_amdgcn_wmma_f32_16x16x4_f32(
                /*neg_a=*/false, a, /*neg_b=*/false, b,
                /*c_mod=*/(short)0, c, /*reuse_a=*/false, /*reuse_b=*/false);
    }

    // ---- D layout: VGPR j holds rows M=j (lanes 0-15) and M=j+8 (16-31)
    const int mb = hi * 8;
#pragma unroll
    for (int j = 0; j < 8; ++j) {
        const int t = t0 + 16 * (mb + j) + nm;
        out[(size_t)e * T + t] = c[j];
    }
}

// ---------------------------------------------------------------------------
extern "C" void kernel_launch(void* const* d_in, const int* in_sizes, int n_in,
                              void* d_out, int out_size, void* d_ws, size_t ws_size,
                              hipStream_t stream) {
    const float* raw     = (const float*)d_in[0];  // [E,29]
    const float* forcing = (const float*)d_in[1];  // [T,3]
    const float* st0     = (const float*)d_in[2];  // [2]
    const float* areaf   = (const float*)d_in[3];  // [NB]
    const float* elev    = (const float*)d_in[4];  // [NB]

    const int E = in_sizes[0] / 29;
    const int T = in_sizes[1] / 3;

    float* runoff = (float*)d_ws;                  // E*T floats (2 MB)

    fuse_scan_kernel<<<(E + 1) / 2, 32, 0, stream>>>(
        raw, forcing, st0, areaf, elev, runoff, T, E);

    const int tilesPerE = T / TILE_T;              // T=8192 -> 32 tiles
    fuse_route_wmma_kernel<<<E * tilesPerE, 32, 0, stream>>>(
        raw, runoff, (float*)d_out, T, tilesPerE);
}